// ProximityAwareLoss3Class_80994493268323
// MI455X (gfx1250) — compile-verified
//
#include <hip/hip_runtime.h>
#include <stdint.h>

// ---------------- problem constants ----------------
#define B_      32
#define S_      65536
#define CHUNK   2048                 // elements of one row per block
#define TPB     256                  // 8 wave32s per block
#define EPT     8                    // elements per thread (CHUNK = TPB*EPT)
#define CPR     (S_ / CHUNK)         // 32 chunks per row
#define NCHUNKS (B_ * CPR)           // 1024 blocks
#define BIGP    (1 << 26)            // position sentinel
#define ITPF    100.0f               // invalid transition penalty
#define TOLI    5                    // proximity tolerance
#define FPF     10.0f                // false positive penalty
#define SWW     30.0f                // switch class weight

// ---------------- workspace layout (bytes) ----------------
#define CE_BYTES    ((size_t)B_ * S_ * 4)        // fp32 CE per element (8 MiB, L2-resident)
#define INFO_BYTES  ((size_t)B_ * S_)            // packed pred|labcode byte  (2 MiB)
#define AGG_BYTES   ((size_t)NCHUNKS * 12 * 4)   // per-chunk summaries
#define SEED_BYTES  ((size_t)NCHUNKS * 10 * 4)   // per-chunk scan seeds
#define ANY_BYTES   ((size_t)B_ * 4 * 4)         // per-row any() flags
#define OFF_INFO    (CE_BYTES)
#define OFF_AGG     (OFF_INFO + INFO_BYTES)
#define OFF_SEED    (OFF_AGG + AGG_BYTES)
#define OFF_ANY     (OFF_SEED + SEED_BYTES)
#define OFF_ACC     (OFF_ANY + ANY_BYTES)

// ---------------- CDNA5 async global->LDS path ----------------
#if defined(__HIP_DEVICE_COMPILE__) && __has_builtin(__builtin_amdgcn_global_load_async_to_lds_b128)
  #define HAVE_ASYNC 1
  #pragma message("CDNA5 path: global_load_async_to_lds_b128 ENABLED")
#else
  #define HAVE_ASYNC 0
  #pragma message("CDNA5 path: async-to-LDS builtin not found; direct LDS staging fallback")
#endif

typedef int v4i_pal __attribute__((vector_size(16)));

__device__ __forceinline__ int imax(int a, int b) { return a > b ? a : b; }
__device__ __forceinline__ int imin(int a, int b) { return a < b ? a : b; }

__device__ __forceinline__ void cdna5_async_copy16(const void* g, void* l) {
#if HAVE_ASYNC
  __builtin_amdgcn_global_load_async_to_lds_b128(
      (__attribute__((address_space(1))) v4i_pal*)(uintptr_t)g,
      (__attribute__((address_space(3))) v4i_pal*)(uintptr_t)l, 0, 0);
#else
  *(float4*)l = *(const float4*)g;
#endif
}

__device__ __forceinline__ void cdna5_async_wait() {
#if HAVE_ASYNC
  #if __has_builtin(__builtin_amdgcn_s_wait_asynccnt)
  __builtin_amdgcn_s_wait_asynccnt(0);
  #else
  asm volatile("s_wait_asynccnt 0" ::: "memory");
  #endif
#endif
}

// ---------------- block reduce / scan helpers ----------------
__device__ __forceinline__ int blkmax(int v, volatile int* sc, int tid) {
  sc[tid] = v; __syncthreads();
  for (int off = TPB / 2; off > 0; off >>= 1) {
    if (tid < off) { int u = sc[tid + off]; if (u > sc[tid]) sc[tid] = u; }
    __syncthreads();
  }
  int r = sc[0]; __syncthreads();
  return r;
}
__device__ __forceinline__ int blkmin(int v, volatile int* sc, int tid) {
  sc[tid] = v; __syncthreads();
  for (int off = TPB / 2; off > 0; off >>= 1) {
    if (tid < off) { int u = sc[tid + off]; if (u < sc[tid]) sc[tid] = u; }
    __syncthreads();
  }
  int r = sc[0]; __syncthreads();
  return r;
}
// exclusive forward max-scan (Hillis-Steele), identity `ident`
__device__ __forceinline__ int exscan_max(int v, volatile int* sc, int tid, int ident) {
  sc[tid] = v; __syncthreads();
  for (int off = 1; off < TPB; off <<= 1) {
    int u = (tid >= off) ? sc[tid - off] : ident;
    __syncthreads();
    if (u > sc[tid]) sc[tid] = u;
    __syncthreads();
  }
  int r = (tid > 0) ? sc[tid - 1] : ident;
  __syncthreads();
  return r;
}
// exclusive backward min-scan (suffix min of threads > tid)
__device__ __forceinline__ int exscan_min_rev(int v, volatile int* sc, int tid, int ident) {
  sc[tid] = v; __syncthreads();
  for (int off = 1; off < TPB; off <<= 1) {
    int u = (tid + off < TPB) ? sc[tid + off] : ident;
    __syncthreads();
    if (u < sc[tid]) sc[tid] = u;
    __syncthreads();
  }
  int r = (tid < TPB - 1) ? sc[tid + 1] : ident;
  __syncthreads();
  return r;
}

// =====================================================================
// Pass 1: stream logits+labels (async to LDS), compute CE/pred, per-chunk
//         summaries (mode exit events, per-mask min/max positions).
// Masks: k=0 pred==1, k=1 pred==2, k=2 label==1, k=3 label==2
// =====================================================================
__global__ void __launch_bounds__(TPB) pal_pass1(const float* __restrict__ logits,
                                                 const int* __restrict__ labels,
                                                 float* __restrict__ ce,
                                                 uint8_t* __restrict__ info,
                                                 int* __restrict__ agg) {
  __shared__ float sLog[CHUNK * 3];   // 24 KiB
  __shared__ int   sLab[CHUNK];       //  8 KiB
  __shared__ int   sc[TPB];

  const int tid   = threadIdx.x;
  const int chunk = blockIdx.x;
  const int row   = chunk / CPR;
  const int cs    = (chunk % CPR) * CHUNK;       // row position of chunk start
  const size_t base = (size_t)row * S_ + cs;

  // ---- stage tile into LDS (async copies, 16B granules) ----
  const char* gLog = (const char*)(logits + base * 3);
  const char* gLab = (const char*)(labels + base);
  #pragma unroll
  for (int i = tid; i < (CHUNK * 3 * 4) / 16; i += TPB)
    cdna5_async_copy16(gLog + (size_t)i * 16, (char*)sLog + (size_t)i * 16);
  #pragma unroll
  for (int i = tid; i < (CHUNK * 4) / 16; i += TPB)
    cdna5_async_copy16(gLab + (size_t)i * 16, (char*)sLab + (size_t)i * 16);
  cdna5_async_wait();
  __syncthreads();

  // ---- per-thread element work ----
  const int idx0 = tid * EPT;
  float cev[EPT];
  uint32_t ib[2] = {0u, 0u};
  int lastEvP = 0, lastEvT = 0;
  int mn[4] = {BIGP, BIGP, BIGP, BIGP};
  int mx[4] = {-BIGP, -BIGP, -BIGP, -BIGP};

  #pragma unroll
  for (int j = 0; j < EPT; ++j) {
    const int idx = idx0 + j;
    const int lab = sLab[idx];
    const float a0 = sLog[idx * 3 + 0];
    const float a1 = sLog[idx * 3 + 1];
    const float a2 = sLog[idx * 3 + 2];
    // argmax, first-max-wins like jnp.argmax
    int pred = 0; float bm = a0;
    if (a1 > bm) { pred = 1; bm = a1; }
    if (a2 > bm) { pred = 2; }
    // log-sum-exp over 3 classes
    const float m   = fmaxf(a0, fmaxf(a1, a2));
    const float lse = m + __logf(__expf(a0 - m) + __expf(a1 - m) + __expf(a2 - m));
    const bool valid = (lab != -100);
    const int  labc  = valid ? lab : 3;
    const float w  = (labc == 1 || labc == 2) ? SWW : 1.0f;
    const float ay = (labc == 1) ? a1 : ((labc == 2) ? a2 : a0);
    cev[j] = valid ? (-w * (ay - lse)) : 0.0f;
    ib[j >> 2] |= ((uint32_t)(pred | (labc << 2))) << ((j & 3) * 8);

    const int g = cs + idx;
    if (pred == 1) { mn[0] = imin(mn[0], g); mx[0] = imax(mx[0], g); }
    if (pred == 2) { mn[1] = imin(mn[1], g); mx[1] = imax(mx[1], g); }
    if (labc == 1) { mn[2] = imin(mn[2], g); mx[2] = imax(mx[2], g); }
    if (labc == 2) { mn[3] = imin(mn[3], g); mx[3] = imax(mx[3], g); }
    if (valid && (pred == 1 || pred == 2)) lastEvP = ((idx + 1) << 2) | pred;
    if (labc == 1 || labc == 2)            lastEvT = ((idx + 1) << 2) | labc;
  }

  // vectorized spill of CE + packed info (L2-resident workspace)
  *(float4*)(ce + base + idx0)     = make_float4(cev[0], cev[1], cev[2], cev[3]);
  *(float4*)(ce + base + idx0 + 4) = make_float4(cev[4], cev[5], cev[6], cev[7]);
  *(uint32_t*)(info + base + idx0)     = ib[0];
  *(uint32_t*)(info + base + idx0 + 4) = ib[1];

  // ---- block reductions -> chunk summary ----
  const int eP  = blkmax(lastEvP, sc, tid);
  const int eT  = blkmax(lastEvT, sc, tid);
  const int n0 = blkmin(mn[0], sc, tid), x0 = blkmax(mx[0], sc, tid);
  const int n1 = blkmin(mn[1], sc, tid), x1 = blkmax(mx[1], sc, tid);
  const int n2 = blkmin(mn[2], sc, tid), x2 = blkmax(mx[2], sc, tid);
  const int n3 = blkmin(mn[3], sc, tid), x3 = blkmax(mx[3], sc, tid);
  if (tid == 0) {
    int* a = agg + chunk * 12;
    a[0] = eP & 3;  // 0 none, 1 -> pm=0, 2 -> pm=1
    a[1] = eT & 3;  // 0 none, 1 -> tm=0, 2 -> tm=1
    a[2] = n0; a[3] = x0; a[4] = n1; a[5] = x1;
    a[6] = n2; a[7] = x2; a[8] = n3; a[9] = x3;
    a[10] = 0; a[11] = 0;
  }
}

// =====================================================================
// Pass 2: tiny per-row prefix/suffix over 32 chunk summaries -> seeds,
//         any() flags, zero accumulators.
// =====================================================================
__global__ void __launch_bounds__(64) pal_pass2(const int* __restrict__ agg,
                                                int* __restrict__ seeds,
                                                int* __restrict__ rowAny,
                                                float* __restrict__ acc) {
  const int tid = threadIdx.x;
  if (tid == 0) { acc[0] = 0.0f; acc[1] = 0.0f; }
  if (tid < B_) {
    const int row = tid;
    int pm = 0, tm = 0;
    int prev[4] = {-BIGP, -BIGP, -BIGP, -BIGP};
    for (int ci = 0; ci < CPR; ++ci) {
      const int c = row * CPR + ci;
      int* s = seeds + c * 10;
      s[0] = pm; s[1] = tm;
      #pragma unroll
      for (int k = 0; k < 4; ++k) s[2 + k] = prev[k];
      const int* a = agg + c * 12;
      if (a[0] == 1) pm = 0; else if (a[0] == 2) pm = 1;
      if (a[1] == 1) tm = 0; else if (a[1] == 2) tm = 1;
      #pragma unroll
      for (int k = 0; k < 4; ++k) prev[k] = imax(prev[k], a[3 + 2 * k]);
    }
    int nxt[4] = {BIGP, BIGP, BIGP, BIGP};
    int any[4] = {0, 0, 0, 0};
    for (int ci = CPR - 1; ci >= 0; --ci) {
      const int c = row * CPR + ci;
      int* s = seeds + c * 10;
      const int* a = agg + c * 12;
      #pragma unroll
      for (int k = 0; k < 4; ++k) {
        s[6 + k] = nxt[k];
        nxt[k] = imin(nxt[k], a[2 + 2 * k]);
        if (a[3 + 2 * k] >= 0) any[k] = 1;
      }
    }
    #pragma unroll
    for (int k = 0; k < 4; ++k) rowAny[row * 4 + k] = any[k];
  }
}

// =====================================================================
// Pass 3: seeded in-chunk scans (mode state, prev/next mask positions),
//         apply all proximity factors, accumulate masked sum + count.
// =====================================================================
__global__ void __launch_bounds__(TPB) pal_pass3(const float* __restrict__ ce,
                                                 const uint8_t* __restrict__ info,
                                                 const int* __restrict__ seeds,
                                                 const int* __restrict__ rowAny,
                                                 float* __restrict__ acc) {
  __shared__ int   sc[TPB];
  __shared__ float sf[TPB];

  const int tid   = threadIdx.x;
  const int chunk = blockIdx.x;
  const int row   = chunk / CPR;
  const int cs    = (chunk % CPR) * CHUNK;
  const size_t base = (size_t)row * S_ + cs;

  __builtin_prefetch(ce + base + (size_t)tid * EPT, 0, 3);    // global_prefetch_b8
  __builtin_prefetch(info + base + (size_t)tid * EPT, 0, 3);

  const int* sd = seeds + chunk * 10;
  const int seedPm = sd[0], seedTm = sd[1];
  const int seedPrev[4] = {sd[2], sd[3], sd[4], sd[5]};
  const int seedNext[4] = {sd[6], sd[7], sd[8], sd[9]};
  const int anyk[4] = {rowAny[row * 4 + 0], rowAny[row * 4 + 1],
                       rowAny[row * 4 + 2], rowAny[row * 4 + 3]};

  const int idx0 = tid * EPT;
  const float4 c0 = *(const float4*)(ce + base + idx0);
  const float4 c1 = *(const float4*)(ce + base + idx0 + 4);
  const float cev[EPT] = {c0.x, c0.y, c0.z, c0.w, c1.x, c1.y, c1.z, c1.w};
  const uint32_t i0 = *(const uint32_t*)(info + base + idx0);
  const uint32_t i1 = *(const uint32_t*)(info + base + idx0 + 4);

  int pr8[EPT], lc8[EPT];
  #pragma unroll
  for (int j = 0; j < EPT; ++j) {
    const uint32_t by = ((j < 4 ? i0 : i1) >> ((j & 3) * 8)) & 0xffu;
    pr8[j] = (int)(by & 3u);
    lc8[j] = (int)((by >> 2) & 3u);
  }

  // per-thread aggregates for the block-wide scans
  int evP = 0, evT = 0;
  int lp[4] = {-BIGP, -BIGP, -BIGP, -BIGP};
  int fp_[4] = {BIGP, BIGP, BIGP, BIGP};
  #pragma unroll
  for (int j = 0; j < EPT; ++j) {
    const int g = cs + idx0 + j;
    const bool mk[4] = {pr8[j] == 1, pr8[j] == 2, lc8[j] == 1, lc8[j] == 2};
    #pragma unroll
    for (int k = 0; k < 4; ++k)
      if (mk[k]) { lp[k] = g; fp_[k] = imin(fp_[k], g); }
    const bool v = (lc8[j] != 3);
    if (v && (pr8[j] == 1 || pr8[j] == 2)) evP = ((idx0 + j + 1) << 2) | pr8[j];
    if (lc8[j] == 1 || lc8[j] == 2)        evT = ((idx0 + j + 1) << 2) | lc8[j];
  }

  // block-wide exclusive scans, seeded with cross-chunk state
  const int xEvP = exscan_max(evP, sc, tid, 0);
  const int xEvT = exscan_max(evT, sc, tid, 0);
  int last[4], nxt[4];
  for (int k = 0; k < 4; ++k)
    last[k] = imax(exscan_max(lp[k], sc, tid, -BIGP), seedPrev[k]);
  for (int k = 0; k < 4; ++k)
    nxt[k] = imin(exscan_min_rev(fp_[k], sc, tid, BIGP), seedNext[k]);
  int pm = xEvP ? (((xEvP & 3) == 1) ? 0 : 1) : seedPm;
  int tm = xEvT ? (((xEvT & 3) == 1) ? 0 : 1) : seedTm;

  // backward sweep: inclusive next-position per element
  int nA[EPT * 4];
  #pragma unroll
  for (int j = EPT - 1; j >= 0; --j) {
    const int g = cs + idx0 + j;
    const bool mk[4] = {pr8[j] == 1, pr8[j] == 2, lc8[j] == 1, lc8[j] == 2};
    #pragma unroll
    for (int k = 0; k < 4; ++k) {
      if (mk[k]) nxt[k] = g;
      nA[j * 4 + k] = nxt[k];
    }
  }

  // forward sweep: mode multiplier + proximity factors + accumulate
  float sum = 0.0f;
  int cnt = 0;
  #pragma unroll
  for (int j = 0; j < EPT; ++j) {
    const int g = cs + idx0 + j;
    const int p = pr8[j], l = lc8[j];
    const bool v = (l != 3);
    const bool mk0 = (p == 1), mk1 = (p == 2), mk2 = (l == 1), mk3 = (l == 2);

    float mm = 1.0f;
    const bool p1 = v && mk0, p2 = v && mk1;
    if (p1 && pm == 0) mm *= ITPF;
    if (p2 && pm == 1) mm *= ITPF;
    if (p1) pm = 0; else if (p2) pm = 1;
    if (mk2 && tm == 1 && p == 1) mm *= 0.1f;
    if (mk3 && tm == 0 && p == 2) mm *= 0.1f;
    if (mk2) tm = 0; else if (mk3) tm = 1;

    if (mk0) last[0] = g; if (mk1) last[1] = g;
    if (mk2) last[2] = g; if (mk3) last[3] = g;

    float adj = cev[j] * mm;
    if (mk0 || mk1) {                 // predicted switch -> distance to same-class truth
      const int k = mk0 ? 2 : 3;
      const int d = imin(g - last[k], nA[j * 4 + k] - g);
      float f;
      if (!anyk[k])      f = FPF * 2.0f;
      else if (d == 0)   f = 0.1f;
      else if (d <= TOLI) f = 0.1f + (float)d * (0.9f / (float)TOLI);
      else               f = FPF;
      adj *= f;
    }
    if (mk2 || mk3) {                 // true switch -> distance to same-class prediction
      const int k = mk2 ? 0 : 1;
      const int d = imin(g - last[k], nA[j * 4 + k] - g);
      float f;
      if (!anyk[k])      f = 5.0f;
      else if (d > TOLI) f = fminf(2.0f + (float)(d - TOLI) * 0.3f, 8.0f);
      else               f = 1.0f;
      adj *= f;
    }
    if (v) { sum += adj; cnt++; }
  }

  // block reduce + atomic accumulate
  sf[tid] = sum; __syncthreads();
  for (int off = TPB / 2; off > 0; off >>= 1) {
    if (tid < off) sf[tid] += sf[tid + off];
    __syncthreads();
  }
  const float bs = sf[0]; __syncthreads();
  sc[tid] = cnt; __syncthreads();
  for (int off = TPB / 2; off > 0; off >>= 1) {
    if (tid < off) sc[tid] += sc[tid + off];
    __syncthreads();
  }
  if (tid == 0) {
    atomicAdd(&acc[0], bs);
    atomicAdd(&acc[1], (float)sc[0]);
  }
}

__global__ void pal_finalize(const float* __restrict__ acc, float* __restrict__ out) {
  if (threadIdx.x == 0 && blockIdx.x == 0)
    out[0] = acc[0] / fmaxf(acc[1], 1.0f);
}

// =====================================================================
extern "C" void kernel_launch(void* const* d_in, const int* in_sizes, int n_in,
                              void* d_out, int out_size, void* d_ws, size_t ws_size,
                              hipStream_t stream) {
  (void)in_sizes; (void)n_in; (void)out_size; (void)ws_size;
  const float* logits = (const float*)d_in[0];
  const int*   labels = (const int*)d_in[1];

  char* ws = (char*)d_ws;
  float*   ce     = (float*)ws;
  uint8_t* info   = (uint8_t*)(ws + OFF_INFO);
  int*     agg    = (int*)(ws + OFF_AGG);
  int*     seeds  = (int*)(ws + OFF_SEED);
  int*     rowAny = (int*)(ws + OFF_ANY);
  float*   acc    = (float*)(ws + OFF_ACC);

  pal_pass1<<<NCHUNKS, TPB, 0, stream>>>(logits, labels, ce, info, agg);
  pal_pass2<<<1, 64, 0, stream>>>(agg, seeds, rowAny, acc);
  pal_pass3<<<NCHUNKS, TPB, 0, stream>>>(ce, info, seeds, rowAny, acc);
  pal_finalize<<<1, 1, 0, stream>>>(acc, (float*)d_out);
}